// LocalSpatialEncoding_59201829208476
// MI455X (gfx1250) — compile-verified
//
#include <hip/hip_runtime.h>

// LocalSpatialEncoding for MI455X (gfx1250, wave32).
// Pipeline: zero-stats -> WMMA stats pass -> finalize mean/rstd -> WMMA output pass.
// The 10-channel contraction is padded to 12 and done with V_WMMA_F32_16X16X4_F32
// (4 M-tiles x 3 K-steps = 12 WMMAs per point); bias folded in via concat col 10 == 1.

typedef __attribute__((ext_vector_type(2))) float v2f;
typedef __attribute__((ext_vector_type(8))) float v8f;

#define NGROUPS 16
#define DCH 64
#define KNN 16
#define LSE_EPS 1e-6f

__device__ __forceinline__ v8f wmma_f32_k4(v2f a, v2f b, v8f c) {
  // D = A(16x4,f32) * B(4x16,f32) + C(16x16,f32)
  return __builtin_amdgcn_wmma_f32_16x16x4_f32(
      /*neg_a=*/false, a, /*neg_b=*/false, b,
      /*c_mod=*/(short)0, c, /*reuse_a=*/false, /*reuse_b=*/false);
}

// A-matrix fragments: a[m][s] covers rows d = 16m + lane%16, cols 4s..4s+3
// (lane half picks the col pair). Col 10 carries conv_b (bias), col 11 = 0.
__device__ __forceinline__ void load_a_frags(const float* __restrict__ conv_w,
                                             const float* __restrict__ conv_b,
                                             int kcol, int h, v2f a[4][3]) {
#pragma unroll
  for (int m = 0; m < 4; ++m) {
#pragma unroll
    for (int s = 0; s < 3; ++s) {
      const int d  = 16 * m + kcol;
      const int c0 = 4 * s + 2 * h;
      const int c1 = c0 + 1;
      a[m][s].x = (c0 < 10) ? conv_w[d * 10 + c0] : ((c0 == 10) ? conv_b[d] : 0.f);
      a[m][s].y = (c1 < 10) ? conv_w[d * 10 + c1] : ((c1 == 10) ? conv_b[d] : 0.f);
    }
  }
}

// B-matrix fragments for one point n: column k = lane%16 holds the 12-entry
// concat vector [ctr, nbr, ctr-nbr, dist, 1, 0]; lane half picks the row pair.
__device__ __forceinline__ void build_b_frags(const float* __restrict__ coords,
                                              const int* __restrict__ idxp,
                                              const float* __restrict__ distp,
                                              int b, int n, int N, int kcol, int h,
                                              v2f bf[3], int* j_out) {
  const size_t pbase = (size_t)b * N + n;
  const int j = idxp[pbase * KNN + kcol];
  const float cx = coords[pbase * 3 + 0];
  const float cy = coords[pbase * 3 + 1];
  const float cz = coords[pbase * 3 + 2];
  const size_t nbase = (size_t)b * N + j;
  const float nx = coords[nbase * 3 + 0];
  const float ny = coords[nbase * 3 + 1];
  const float nz = coords[nbase * 3 + 2];
  const float dd = distp[pbase * KNN + kcol];
  const float cc[12] = {cx, cy, cz, nx, ny, nz, cx - nx, cy - ny, cz - nz, dd, 1.0f, 0.0f};
#pragma unroll
  for (int s = 0; s < 3; ++s) {  // constant indices only -> stays in VGPRs
    bf[s].x = h ? cc[4 * s + 2] : cc[4 * s + 0];
    bf[s].y = h ? cc[4 * s + 3] : cc[4 * s + 1];
  }
  *j_out = j;
}

__device__ __forceinline__ void mma_tile(const v2f a[4][3], const v2f bf[3], v8f acc[4]) {
#pragma unroll
  for (int m = 0; m < 4; ++m) {
    acc[m] = (v8f){0.f, 0.f, 0.f, 0.f, 0.f, 0.f, 0.f, 0.f};
#pragma unroll
    for (int s = 0; s < 3; ++s) acc[m] = wmma_f32_k4(a[m][s], bf[s], acc[m]);
  }
}

__global__ void lse_init_kernel(float* __restrict__ stats, int n) {
  const int t = blockIdx.x * blockDim.x + threadIdx.x;
  if (t < n) stats[t] = 0.f;
}

// ---------------- Pass 1: per-group sum / sumsq ----------------
__global__ __launch_bounds__(256) void lse_stats_kernel(
    const float* __restrict__ coords, const int* __restrict__ idxp,
    const float* __restrict__ distp, const float* __restrict__ conv_w,
    const float* __restrict__ conv_b, float* __restrict__ stats, int N) {
  const int b    = blockIdx.y;
  const int lane = threadIdx.x & 31;
  const int kcol = lane & 15;
  const int h    = lane >> 4;
  const int wavesPerB = gridDim.x * (blockDim.x >> 5);
  const int wid       = blockIdx.x * (blockDim.x >> 5) + (threadIdx.x >> 5);

  v2f a[4][3];
  load_a_frags(conv_w, conv_b, kcol, h, a);

  float s_sum[8], s_sq[8];
#pragma unroll
  for (int i = 0; i < 8; ++i) { s_sum[i] = 0.f; s_sq[i] = 0.f; }

  for (int n = wid; n < N; n += wavesPerB) {   // uniform per wave -> EXEC all 1s at WMMA
    v2f bf[3]; int j;
    build_b_frags(coords, idxp, distp, b, n, N, kcol, h, bf, &j);
    v8f acc[4];
    mma_tile(a, bf, acc);
#pragma unroll
    for (int m = 0; m < 4; ++m) {
#pragma unroll
      for (int r = 0; r < 8; ++r) {
        const float xv = acc[m][r];         // channel d = 16m + r + 8h, k = kcol
        const int li = 2 * m + (r >> 2);    // compile-time slot; group = f(li, h)
        s_sum[li] += xv;
        s_sq[li]  += xv * xv;
      }
    }
  }

#pragma unroll
  for (int li = 0; li < 8; ++li) {
    float s = s_sum[li], q = s_sq[li];
#pragma unroll
    for (int mask = 8; mask >= 1; mask >>= 1) {  // half-wave reduce (halves hold different groups)
      s += __shfl_xor(s, mask, 32);
      q += __shfl_xor(q, mask, 32);
    }
    if (kcol == 0) {
      const int g = 4 * (li >> 1) + (li & 1) + 2 * h;
      atomicAdd(&stats[(b * NGROUPS + g) * 2 + 0], s);
      atomicAdd(&stats[(b * NGROUPS + g) * 2 + 1], q);
    }
  }
}

__global__ void lse_finalize_kernel(const float* __restrict__ stats,
                                    float* __restrict__ ms, float inv_cnt, int n) {
  const int t = blockIdx.x * blockDim.x + threadIdx.x;
  if (t < n) {
    const float s = stats[2 * t], q = stats[2 * t + 1];
    const float mean = s * inv_cnt;
    const float var  = fmaf(-mean, mean, q * inv_cnt);
    ms[2 * t]     = mean;
    ms[2 * t + 1] = rsqrtf(var + LSE_EPS);
  }
}

// ---------------- Pass 2: normalize + ReLU + feature gather + store ----------------
__global__ __launch_bounds__(256) void lse_out_kernel(
    const float* __restrict__ coords, const float* __restrict__ features,
    const int* __restrict__ idxp, const float* __restrict__ distp,
    const float* __restrict__ conv_w, const float* __restrict__ conv_b,
    const float* __restrict__ gamma, const float* __restrict__ beta,
    const float* __restrict__ ms, float* __restrict__ out, int N) {
  const int b    = blockIdx.y;
  const int lane = threadIdx.x & 31;
  const int kcol = lane & 15;
  const int h    = lane >> 4;
  const int wavesPerB = gridDim.x * (blockDim.x >> 5);
  const int wid       = blockIdx.x * (blockDim.x >> 5) + (threadIdx.x >> 5);

  v2f a[4][3];
  load_a_frags(conv_w, conv_b, kcol, h, a);

  // Fold GroupNorm affine into per-(m,r) scale/shift: y = x*scal + shft
  float scal[4][8], shft[4][8];
#pragma unroll
  for (int m = 0; m < 4; ++m) {
#pragma unroll
    for (int r = 0; r < 8; ++r) {
      const int d  = 16 * m + r + 8 * h;
      const int g  = d >> 2;
      const float mean = ms[(b * NGROUPS + g) * 2 + 0];
      const float rstd = ms[(b * NGROUPS + g) * 2 + 1];
      const float sc = rstd * gamma[d];
      scal[m][r] = sc;
      shft[m][r] = fmaf(-mean, sc, beta[d]);
    }
  }

  for (int n = wid; n < N; n += wavesPerB) {
    v2f bf[3]; int j;
    build_b_frags(coords, idxp, distp, b, n, N, kcol, h, bf, &j);
    v8f acc[4];
    mma_tile(a, bf, acc);
    // x half of the output: each half-wave store is a contiguous 64B row
#pragma unroll
    for (int m = 0; m < 4; ++m) {
#pragma unroll
      for (int r = 0; r < 8; ++r) {
        const int d = 16 * m + r + 8 * h;
        float y = fmaf(acc[m][r], scal[m][r], shft[m][r]);
        y = fmaxf(y, 0.f);
        out[((size_t)(b * 2 * DCH + d) * N + n) * KNN + kcol] = y;
      }
    }
    // neighbor-feature gather (features table fits in L2); two d rows per iter
#pragma unroll 8
    for (int dd = 0; dd < 32; ++dd) {
      const int d = 2 * dd + h;
      const float fv = features[(size_t)(b * DCH + d) * N + j];
      out[((size_t)(b * 2 * DCH + DCH + d) * N + n) * KNN + kcol] = fv;
    }
  }
}

extern "C" void kernel_launch(void* const* d_in, const int* in_sizes, int n_in,
                              void* d_out, int out_size, void* d_ws, size_t ws_size,
                              hipStream_t stream) {
  (void)n_in; (void)out_size; (void)ws_size;
  const float* coords   = (const float*)d_in[0];
  const float* features = (const float*)d_in[1];
  const int*   idxp     = (const int*)d_in[2];
  const float* distp    = (const float*)d_in[3];
  const float* conv_w   = (const float*)d_in[4];
  const float* conv_b   = (const float*)d_in[5];
  const float* gamma    = (const float*)d_in[6];
  const float* beta     = (const float*)d_in[7];

  const int B = 2;
  const int N = in_sizes[0] / (3 * B);

  float* stats = (float*)d_ws;              // [B*16][2] sum/sumsq
  float* ms    = stats + 2 * B * NGROUPS;   // [B*16][2] mean/rstd

  lse_init_kernel<<<1, 64, 0, stream>>>(stats, B * NGROUPS * 2);

  dim3 g1(256, B);
  lse_stats_kernel<<<g1, 256, 0, stream>>>(coords, idxp, distp, conv_w, conv_b, stats, N);

  const float inv_cnt = 1.0f / ((float)(DCH / NGROUPS) * (float)N * (float)KNN);
  lse_finalize_kernel<<<1, 64, 0, stream>>>(stats, ms, inv_cnt, B * NGROUPS);

  dim3 g2(512, B);
  lse_out_kernel<<<g2, 256, 0, stream>>>(coords, features, idxp, distp, conv_w, conv_b,
                                         gamma, beta, ms, (float*)d_out, N);
}